// DensityPropCNN_89541478187438
// MI455X (gfx1250) — compile-verified
//
#include <hip/hip_runtime.h>
#include <hip/hip_bf16.h>
#include <math.h>

// ---------------------------------------------------------------------------
// DensityPropCNN on gfx1250 (MI455X): fp32 WMMA (V_WMMA_F32_16X16X4_F32)
// Rank-25 factorization avoids materializing [B,16,576,576] sigma tensors.
// k_z stages its GEMM tiles into LDS via GLOBAL_LOAD_ASYNC_TO_LDS_B128
// (ASYNCcnt path) so WMMA fragments come from LDS instead of strided global.
// ---------------------------------------------------------------------------

typedef __attribute__((ext_vector_type(2))) float v2f;
typedef __attribute__((ext_vector_type(8))) float v8f;

#define BATCH 16
#define NKER  16
#define KSZ   5
#define PATCH 25
#define CPAD  28      // 25 patch elems padded to 28 (7 K-steps of 4)
#define HOUT  24
#define POUT  576     // 24*24
#define HP    144     // 12*12 pooled pixels
#define UNITS 10

// workspace layout (float offsets)
#define OFF_XM   0                                    // [B][576][28] patches
#define OFF_MU1  (OFF_XM  + BATCH*POUT*CPAD)          // [B][576][16]
#define OFF_MU3  (OFF_MU1 + BATCH*POUT*NKER)          // [B][16][144] == mu_flatt
#define OFF_Y    (OFF_MU3 + BATCH*NKER*HP)            // [B*16][144][28] sigma factor
#define OFF_Z    (OFF_Y   + BATCH*NKER*HP*CPAD)       // [B*16][16][32]
#define OFF_SIGA (OFF_Z   + BATCH*NKER*16*32)         // [B][16][16]
#define OFF_MU4  (OFF_SIGA+ BATCH*256)                // [16][16]
#define OFF_TRM  (OFF_MU4 + 256)                      // [B][2] : (trace, mnorm)

__device__ __forceinline__ float cleanf(float v) {
  return (__builtin_isnan(v) || __builtin_isinf(v)) ? 0.0f : v;
}

// D(16x16,f32) = A(16x4,f32) * B(4x16,f32) + C
__device__ __forceinline__ v8f wmma4(v2f a, v2f b, v8f c) {
  return __builtin_amdgcn_wmma_f32_16x16x4_f32(false, a, false, b,
                                               (short)0, c, false, false);
}

// gfx1250 async copy: 16B global -> LDS, tracked by ASYNCcnt
__device__ __forceinline__ void async_copy_b128(void* lds_dst, const void* gsrc) {
  uint32_t ldsoff = (uint32_t)(uintptr_t)lds_dst;  // flat->LDS aperture: low 32b
  asm volatile("global_load_async_to_lds_b128 %0, %1, off"
               :: "v"(ldsoff), "v"(gsrc) : "memory");
}
__device__ __forceinline__ void async_wait0() {
  asm volatile("s_wait_asynccnt 0x0" ::: "memory");
}

// --- Kernel 1: im2col patches (c = kh*5+kw ordering), zero-padded to 28 ---
__global__ void k_patches(const float* __restrict__ x, float* __restrict__ ws) {
  __shared__ float xs[28 * 28];
  const int b = blockIdx.x;
  for (int t = threadIdx.x; t < 28 * 28; t += blockDim.x) xs[t] = x[b * 784 + t];
  __syncthreads();
  float* xm = ws + OFF_XM + (size_t)b * POUT * CPAD;
  for (int n = threadIdx.x; n < POUT * CPAD; n += blockDim.x) {
    int p = n / CPAD, c = n % CPAD;
    float v = 0.0f;
    if (c < PATCH) {
      int kh = c / KSZ, kw = c % KSZ;
      int ph = p / HOUT, pw = p % HOUT;
      v = xs[(ph + kh) * 28 + (pw + kw)];
    }
    xm[n] = v;
  }
}

// --- Kernel 2: conv as GEMM  mu1[p,k] = xm[576x28] @ Wt[28x16]  (1 wave/b) ---
__global__ void k_conv(const float* __restrict__ cw, float* __restrict__ ws) {
  const int b    = blockIdx.x;
  const int lane = threadIdx.x;               // 32 threads
  const int half = (lane < 16) ? 0 : 2;
  const int n    = lane & 15;
  const float* xm  = ws + OFF_XM  + (size_t)b * POUT * CPAD;
  float*       mu1 = ws + OFF_MU1 + (size_t)b * POUT * NKER;
  v2f bf[7];                                  // Wt[c,k] = conv_w_mu[k*25+c]
  for (int s = 0; s < 7; ++s) {
    int r0 = 4 * s + half;
    bf[s].x = (r0     < PATCH) ? cw[n * PATCH + r0]     : 0.0f;
    bf[s].y = (r0 + 1 < PATCH) ? cw[n * PATCH + r0 + 1] : 0.0f;
  }
  for (int mt = 0; mt < 36; ++mt) {
    const int mb = mt * 16;
    v8f acc; for (int r = 0; r < 8; ++r) acc[r] = 0.0f;
    const float* arow = xm + (size_t)(mb + n) * CPAD;
    for (int s = 0; s < 7; ++s) {
      int c0 = 4 * s + half;
      v2f a; a.x = arow[c0]; a.y = arow[c0 + 1];
      acc = wmma4(a, bf[s], acc);
    }
    for (int r = 0; r < 8; ++r) {
      int row = mb + r + ((lane < 16) ? 0 : 8);
      mu1[row * NKER + n] = acc[r];
    }
  }
}

// --- Kernel 3: relu+maxpool argmax; emit y[i,:] = g*sqrt(softplus)*patch ----
__global__ void k_pool(const float* __restrict__ csig, float* __restrict__ ws) {
  const int bk = blockIdx.x;                  // b*16 + k
  const int b = bk >> 4, k = bk & 15;
  const int i = threadIdx.x;
  if (i >= HP) return;
  const float* mu1 = ws + OFF_MU1 + (size_t)b * POUT * NKER;
  const int ih = i / 12, iw = i % 12;
  const int h0 = 2 * ih, w0 = 2 * iw;
  float v00 = mu1[(h0 * HOUT + w0) * NKER + k];
  float v01 = mu1[(h0 * HOUT + w0 + 1) * NKER + k];
  float v10 = mu1[((h0 + 1) * HOUT + w0) * NKER + k];
  float v11 = mu1[((h0 + 1) * HOUT + w0 + 1) * NKER + k];
  float m = v00; int a = 0;                   // first-occurrence argmax
  if (v01 > m) { m = v01; a = 1; }
  if (v10 > m) { m = v10; a = 2; }
  if (v11 > m) { m = v11; a = 3; }
  ws[OFF_MU3 + (size_t)bk * HP + i] = fmaxf(m, 0.0f);
  const int   idx   = (h0 + (a >> 1)) * HOUT + (w0 + (a & 1));
  const float scale = (m > 0.0f) ? sqrtf(log1pf(expf(csig[k]))) : 0.0f;
  const float* src = ws + OFF_XM + (size_t)b * POUT * CPAD + (size_t)idx * CPAD;
  float*       dst = ws + OFF_Y + ((size_t)bk * HP + i) * CPAD;
  for (int c = 0; c < CPAD; ++c) dst[c] = scale * src[c];
}

// --- Kernel 4: Z[b,k] = fcw[k]^T (10x144) @ y (144x28); 2 waves = 2 N-tiles.
//     Tiles staged global->LDS via async (ASYNCcnt), fragments read from LDS.
__global__ void k_z(const float* __restrict__ fcw, float* __restrict__ ws) {
  __shared__ __align__(16) float ys[HP * CPAD];   // 16128 B
  __shared__ __align__(16) float wk[HP * UNITS];  //  5760 B
  const int bk  = blockIdx.x;
  const int k   = bk & 15;
  const int tid = threadIdx.x;                    // 64 threads = 2 waves
  const float* ysrc = ws + OFF_Y + (size_t)bk * HP * CPAD;
  const float* wsrc = fcw + (size_t)k * HP * UNITS;
  for (int t = tid; t < (HP * CPAD) / 4; t += 64)
    async_copy_b128(&ys[t * 4], ysrc + t * 4);
  for (int t = tid; t < (HP * UNITS) / 4; t += 64)
    async_copy_b128(&wk[t * 4], wsrc + t * 4);
  async_wait0();
  __syncthreads();

  const int wave = tid >> 5;
  const int lane = tid & 31;
  const int half = (lane < 16) ? 0 : 2;
  const int n    = lane & 15;
  const int c    = wave * 16 + n;                 // output column (0..31)
  v8f acc; for (int r = 0; r < 8; ++r) acc[r] = 0.0f;
  for (int s = 0; s < 36; ++s) {
    int i0 = 4 * s + half;
    v2f a, bb;                                    // A[u,i] = fcw[k][i][u]
    a.x  = (n < UNITS) ? wk[i0 * UNITS + n]       : 0.0f;
    a.y  = (n < UNITS) ? wk[(i0 + 1) * UNITS + n] : 0.0f;
    bb.x = (c < CPAD) ? ys[i0 * CPAD + c]         : 0.0f;
    bb.y = (c < CPAD) ? ys[(i0 + 1) * CPAD + c]   : 0.0f;
    acc = wmma4(a, bb, acc);
  }
  float* z = ws + OFF_Z + (size_t)bk * 16 * 32;
  for (int r = 0; r < 8; ++r) {
    int u = r + ((lane < 16) ? 0 : 8);
    z[u * 32 + wave * 16 + n] = acc[r];
  }
}

// --- Kernel 5: sigma_a[b] = sum_k Z Z^T (Gram: A-frag == B-frag); tr,mnorm --
__global__ void k_siga(float* __restrict__ ws) {
  const int b    = blockIdx.x;
  const int lane = threadIdx.x;               // 32 threads
  const int half = (lane < 16) ? 0 : 2;
  const int n    = lane & 15;
  v8f acc; for (int r = 0; r < 8; ++r) acc[r] = 0.0f;
  for (int k = 0; k < NKER; ++k) {
    const float* z = ws + OFF_Z + (size_t)(b * NKER + k) * 16 * 32;
    if (k + 1 < NKER)                          // global_prefetch_b8 next tile
      __builtin_prefetch(ws + OFF_Z + (size_t)(b * NKER + k + 1) * 16 * 32, 0, 0);
    for (int s = 0; s < 8; ++s) {
      int c0 = 4 * s + half;
      v2f a; a.x = z[n * 32 + c0]; a.y = z[n * 32 + c0 + 1];
      acc = wmma4(a, a, acc);                 // Z @ Z^T : symmetric fragments
    }
  }
  float* siga = ws + OFF_SIGA + (size_t)b * 256;
  for (int r = 0; r < 8; ++r) siga[(r + ((lane < 16) ? 0 : 8)) * 16 + n] = acc[r];
  // trace = sum |y|^2 ; mnorm = sum mu3^2 (wave reductions, deterministic)
  const float* yb = ws + OFF_Y + (size_t)b * NKER * HP * CPAD;
  float t = 0.0f;
  for (int j = lane; j < NKER * HP * CPAD; j += 32) { float v = yb[j]; t = fmaf(v, v, t); }
  for (int o = 16; o > 0; o >>= 1) t += __shfl_xor(t, o, 32);
  const float* m3 = ws + OFF_MU3 + (size_t)b * NKER * HP;
  float q = 0.0f;
  for (int j = lane; j < NKER * HP; j += 32) { float v = m3[j]; q = fmaf(v, v, q); }
  for (int o = 16; o > 0; o >>= 1) q += __shfl_xor(q, o, 32);
  if (lane == 0) { ws[OFF_TRM + b * 2] = t; ws[OFF_TRM + b * 2 + 1] = q; }
}

// --- Kernel 6: mu4 = mu_flatt[16x2304] @ fc_w[2304x10] : one wave, 576 steps -
__global__ void k_mu4(const float* __restrict__ fcw, float* __restrict__ ws) {
  const int lane = threadIdx.x;
  const int half = (lane < 16) ? 0 : 2;
  const int n    = lane & 15;
  const float* m3 = ws + OFF_MU3;             // [16][2304]
  v8f acc; for (int r = 0; r < 8; ++r) acc[r] = 0.0f;
  for (int s = 0; s < 576; ++s) {
    int c0 = 4 * s + half;
    v2f a, bb;
    a.x  = m3[n * 2304 + c0];
    a.y  = m3[n * 2304 + c0 + 1];
    bb.x = (n < UNITS) ? fcw[c0 * UNITS + n]       : 0.0f;
    bb.y = (n < UNITS) ? fcw[(c0 + 1) * UNITS + n] : 0.0f;
    acc = wmma4(a, bb, acc);
  }
  for (int r = 0; r < 8; ++r)
    ws[OFF_MU4 + (r + ((lane < 16) ? 0 : 8)) * 16 + n] = acc[r];
}

// --- Kernel 7: softmax + diag terms + G Sigma G^T sandwich + outputs --------
__global__ void k_out(const float* __restrict__ fcsig, const float* __restrict__ ws,
                      float* __restrict__ out) {
  __shared__ float p[UNITS];
  __shared__ float s4[UNITS][UNITS];
  const int b = blockIdx.x;
  const int tid = threadIdx.x;
  if (tid == 0) {
    float mx = -1e30f;
    for (int u = 0; u < UNITS; ++u) mx = fmaxf(mx, ws[OFF_MU4 + b * 16 + u]);
    float se = 0.0f;
    for (int u = 0; u < UNITS; ++u) { float e = expf(ws[OFF_MU4 + b * 16 + u] - mx); p[u] = e; se += e; }
    for (int u = 0; u < UNITS; ++u) p[u] /= se;
  }
  __syncthreads();
  const float trpm = ws[OFF_TRM + b * 2] + ws[OFF_TRM + b * 2 + 1];
  if (tid < UNITS * UNITS) {
    int u = tid / UNITS, v = tid % UNITS;
    float s = ws[OFF_SIGA + b * 256 + u * 16 + v];
    if (u == v) s += log1pf(expf(fcsig[u])) * trpm;
    s4[u][v] = cleanf(s);
  }
  __syncthreads();
  if (tid < UNITS * UNITS) {
    int i = tid / UNITS, l = tid % UNITS;
    float accv = 0.0f;
    for (int j = 0; j < UNITS; ++j) {
      float gij = p[i] * (((i == j) ? 1.0f : 0.0f) - p[j]);
      float inner = 0.0f;
      for (int m = 0; m < UNITS; ++m) {
        float glm = p[l] * (((l == m) ? 1.0f : 0.0f) - p[m]);
        inner = fmaf(s4[j][m], glm, inner);
      }
      accv = fmaf(gij, inner, accv);
    }
    out[BATCH * UNITS + b * 100 + tid] = cleanf(accv);
  }
  if (tid < UNITS) out[b * UNITS + tid] = p[tid];
}

extern "C" void kernel_launch(void* const* d_in, const int* in_sizes, int n_in,
                              void* d_out, int out_size, void* d_ws, size_t ws_size,
                              hipStream_t stream) {
  (void)in_sizes; (void)n_in; (void)out_size; (void)ws_size;
  const float* x  = (const float*)d_in[0];   // [16,1,28,28]
  const float* cw = (const float*)d_in[1];   // [16,1,5,5]
  const float* cs = (const float*)d_in[2];   // [16]
  const float* fw = (const float*)d_in[3];   // [2304,10]
  const float* fs = (const float*)d_in[4];   // [10]
  float* ws  = (float*)d_ws;
  float* out = (float*)d_out;

  hipLaunchKernelGGL(k_patches, dim3(BATCH),        dim3(256), 0, stream, x,  ws);
  hipLaunchKernelGGL(k_conv,    dim3(BATCH),        dim3(32),  0, stream, cw, ws);
  hipLaunchKernelGGL(k_pool,    dim3(BATCH * NKER), dim3(160), 0, stream, cs, ws);
  hipLaunchKernelGGL(k_z,       dim3(BATCH * NKER), dim3(64),  0, stream, fw, ws);
  hipLaunchKernelGGL(k_siga,    dim3(BATCH),        dim3(32),  0, stream, ws);
  hipLaunchKernelGGL(k_mu4,     dim3(1),            dim3(32),  0, stream, fw, ws);
  hipLaunchKernelGGL(k_out,     dim3(BATCH),        dim3(128), 0, stream, fs, ws, out);
}